// GCN_53137335386623
// MI455X (gfx1250) — compile-verified
//
#include <hip/hip_runtime.h>
#include <math.h>

// GCN 2-layer forward for MI455X (gfx1250, wave32).
// GEMM via V_WMMA_F32_16X16X4_F32; gather/scatter via L2-resident
// global_atomic_add_f32 (51MB agg/h buffers fit the 192MB L2).

#define DCH 128
#define BN_EPS 1e-5f

typedef __attribute__((ext_vector_type(2))) float v2f;
typedef __attribute__((ext_vector_type(8))) float v8f;

// ---------------- degree / normalization ----------------

__global__ void deg_init(float* deg, int n) {
    int i = blockIdx.x * blockDim.x + threadIdx.x;
    if (i < n) deg[i] = 1.0f;   // self loop
}

__global__ void deg_count(const int* __restrict__ dst, float* deg, int E) {
    int e = blockIdx.x * blockDim.x + threadIdx.x;
    if (e < E) atomicAdd(&deg[dst[e]], 1.0f);
}

__global__ void deg_rsqrt(float* deg, int n) {
    int i = blockIdx.x * blockDim.x + threadIdx.x;
    if (i < n) deg[i] = rsqrtf(deg[i]);   // deg >= 1 always (self loop)
}

// ---------------- fp32 WMMA GEMM: H[N,128] = X[N,128] @ W[128,128] ----------------
// One wave = one 16x16 output tile. Block = 8 waves = all 8 column tiles.
// Grid.x = ceil(N/16) row slabs. Address-clamped loads keep EXEC all-1s for
// WMMA even on a ragged tail; full tiles take a branch-free store epilogue.

__global__ __launch_bounds__(256) void gemm_wmma_f32(const float* __restrict__ X,
                                                     const float* __restrict__ W,
                                                     float* __restrict__ H, int n) {
    const int wave = threadIdx.x >> 5;
    const int lane = threadIdx.x & 31;
    const int half = lane >> 4;       // 0: K={0,1} / rows 0-7, 1: K={2,3} / rows 8-15
    const int l16  = lane & 15;

    const int r0 = blockIdx.x * 16;
    const int c0 = wave * 16;

    int arow = r0 + l16;
    if (arow >= n) arow = n - 1;                 // clamp address, keep EXEC full
    const int bcol = c0 + l16;

    const float* xrow = X + (size_t)arow * DCH + half * 2;
    const float* wcol = W + (size_t)(half * 2) * DCH + bcol;

    v8f acc = {};
#pragma unroll 4
    for (int k = 0; k < DCH; k += 4) {
        v2f a = *(const v2f*)(xrow + k);         // A: K=k+2*half, k+1+2*half
        v2f b;
        b.x = wcol[(size_t)k * DCH];             // B: K=k+2*half, col bcol
        b.y = wcol[(size_t)(k + 1) * DCH];       //    K=k+1+2*half
        acc = __builtin_amdgcn_wmma_f32_16x16x4_f32(
            /*neg_a=*/false, a, /*neg_b=*/false, b,
            /*c_mod=*/(short)0, acc, /*reuse_a=*/false, /*reuse_b=*/false);
    }

    const int rbase = r0 + half * 8;             // C/D layout: VGPR v -> row rbase+v
    float* hp = H + (size_t)rbase * DCH + bcol;
    if (r0 + 16 <= n) {
        // full tile: branch-free, 8 coalesced 64B row stores per lane-half
#pragma unroll
        for (int v = 0; v < 8; ++v) hp[(size_t)v * DCH] = acc[v];
    } else {
#pragma unroll
        for (int v = 0; v < 8; ++v)
            if (rbase + v < n) hp[(size_t)v * DCH] = acc[v];
    }
}

// ---------------- agg init: self-loop term + bias ----------------

__global__ void agg_init(const float* __restrict__ H, const float* __restrict__ dinv,
                         const float* __restrict__ bias, float* __restrict__ agg,
                         int total) {
    int t = blockIdx.x * blockDim.x + threadIdx.x;
    if (t >= total) return;
    int i = t >> 7, c = t & (DCH - 1);
    float di = dinv[i];
    agg[t] = H[t] * di * di + bias[c];
}

// ---------------- edge scatter: agg[dst] += h[src] * dinv[src]*dinv[dst] ----------------
// One wave per edge; lane handles 4 channels (float4 gather, 4 atomics).

__global__ void edge_scatter(const float* __restrict__ H, const float* __restrict__ dinv,
                             const int* __restrict__ src, const int* __restrict__ dst,
                             float* __restrict__ agg, int E) {
    long long t = (long long)blockIdx.x * blockDim.x + threadIdx.x;
    int e = (int)(t >> 5);
    if (e >= E) return;
    int q = ((int)t & 31) * 4;
    int s = src[e], d = dst[e];
    float nrm = dinv[s] * dinv[d];
    float4 hv = *(const float4*)(H + (size_t)s * DCH + q);
    float* ap = agg + (size_t)d * DCH + q;
    atomicAdd(ap + 0, hv.x * nrm);
    atomicAdd(ap + 1, hv.y * nrm);
    atomicAdd(ap + 2, hv.z * nrm);
    atomicAdd(ap + 3, hv.w * nrm);
}

// ---------------- batchnorm ----------------

__global__ void stats_zero(float* stats) {
    stats[threadIdx.x] = 0.0f;   // 256 threads: sum[128] + sumsq[128]
}

__global__ void bn_reduce(const float* __restrict__ agg, float* stats, int n) {
    int c = threadIdx.x;                              // 128 threads = channels
    int rows = (n + gridDim.x - 1) / gridDim.x;
    int r0 = blockIdx.x * rows;
    int r1 = r0 + rows; if (r1 > n) r1 = n;
    float s = 0.0f, s2 = 0.0f;
    for (int r = r0; r < r1; ++r) {
        float v = agg[(size_t)r * DCH + c];
        s += v; s2 += v * v;
    }
    atomicAdd(&stats[c], s);
    atomicAdd(&stats[DCH + c], s2);
}

__global__ void bn_finalize(float* stats, int n) {
    int c = threadIdx.x;                              // 128 threads
    float inv_n = 1.0f / (float)n;
    float mu = stats[c] * inv_n;
    float var = stats[DCH + c] * inv_n - mu * mu;
    stats[2 * DCH + c] = mu;
    stats[3 * DCH + c] = rsqrtf(var + BN_EPS);
}

__global__ void bn_apply_relu(float* __restrict__ agg, const float* __restrict__ stats,
                              const float* __restrict__ gamma, const float* __restrict__ beta,
                              int total) {
    int t = blockIdx.x * blockDim.x + threadIdx.x;
    if (t >= total) return;
    int c = t & (DCH - 1);
    float v = (agg[t] - stats[2 * DCH + c]) * stats[3 * DCH + c];
    v = gamma[c] * v + beta[c];
    agg[t] = v > 0.0f ? v : 0.0f;
}

// ---------------- driver ----------------

static void run_layer(const float* X, const float* W, const float* b,
                      const float* gamma, const float* beta,
                      const int* src, const int* dst,
                      const float* dinv, float* h, float* stats,
                      float* out, int n, int E, hipStream_t stream) {
    const int total = n * DCH;
    gemm_wmma_f32<<<(n + 15) / 16, 256, 0, stream>>>(X, W, h, n);
    agg_init<<<(total + 255) / 256, 256, 0, stream>>>(h, dinv, b, out, total);
    long long sc_threads = (long long)E * 32;
    edge_scatter<<<(unsigned)((sc_threads + 255) / 256), 256, 0, stream>>>(
        h, dinv, src, dst, out, E);
    stats_zero<<<1, 256, 0, stream>>>(stats);
    bn_reduce<<<1024, DCH, 0, stream>>>(out, stats, n);
    bn_finalize<<<1, DCH, 0, stream>>>(stats, n);
    bn_apply_relu<<<(total + 255) / 256, 256, 0, stream>>>(out, stats, gamma, beta, total);
}

extern "C" void kernel_launch(void* const* d_in, const int* in_sizes, int n_in,
                              void* d_out, int out_size, void* d_ws, size_t ws_size,
                              hipStream_t stream) {
    const float* x      = (const float*)d_in[0];
    const int*   ei     = (const int*)d_in[1];   // [2, E] row-major
    const float* W1     = (const float*)d_in[2];
    const float* b1     = (const float*)d_in[3];
    const float* gamma1 = (const float*)d_in[4];
    const float* beta1  = (const float*)d_in[5];
    const float* W2     = (const float*)d_in[6];
    const float* b2     = (const float*)d_in[7];
    const float* gamma2 = (const float*)d_in[8];
    const float* beta2  = (const float*)d_in[9];

    const int n = in_sizes[0] / DCH;      // 100000
    const int E = in_sizes[1] / 2;        // 1600000
    const int* src = ei;
    const int* dst = ei + E;

    // workspace: dinv[n] | h[n*DCH] | stats[4*DCH]
    float* dinv  = (float*)d_ws;
    float* h     = dinv + n;              // n*4 bytes is 16B-aligned for n=100000
    float* stats = h + (size_t)n * DCH;

    float* o1 = (float*)d_out;
    float* o2 = o1 + (size_t)n * DCH;

    // symmetric GCN normalization coefficients
    deg_init<<<(n + 255) / 256, 256, 0, stream>>>(dinv, n);
    deg_count<<<(E + 255) / 256, 256, 0, stream>>>(dst, dinv, E);
    deg_rsqrt<<<(n + 255) / 256, 256, 0, stream>>>(dinv, n);

    run_layer(x,  W1, b1, gamma1, beta1, src, dst, dinv, h, stats, o1, n, E, stream);
    run_layer(o1, W2, b2, gamma2, beta2, src, dst, dinv, h, stats, o2, n, E, stream);
}